// TAGModelPaper_893353198377
// MI455X (gfx1250) — compile-verified
//
#include <hip/hip_runtime.h>

// ---------------------------------------------------------------------------
// TAGConv x2 for MI455X (gfx1250, wave32).
// Sparse propagation: float4 gather + global_atomic_add_f32 scatter (L2-resident).
// Dense combine: V_WMMA_F32_16X16X4_F32, one 16-row tile per wave, W_k in LDS,
// software-pipelined fragment loads (prefetch next K-step before current WMMA).
// ---------------------------------------------------------------------------

typedef __attribute__((ext_vector_type(2))) float v2f;
typedef __attribute__((ext_vector_type(8))) float v8f;

#define NFEAT_IN 128

// ----------------------------- small utility kernels -----------------------

__global__ __launch_bounds__(256) void zero_f32(float* __restrict__ p, long long n) {
    long long i = (long long)blockIdx.x * 256 + threadIdx.x;
    if (i < n) p[i] = 0.0f;
}

__global__ __launch_bounds__(256) void deg_accum(const int* __restrict__ dst,
                                                 float* __restrict__ deg, int E) {
    int e = blockIdx.x * 256 + threadIdx.x;
    if (e < E) atomicAdd(&deg[dst[e]], 1.0f);
}

__global__ __launch_bounds__(256) void deg_inv_sqrt(float* __restrict__ deg, int N) {
    int i = blockIdx.x * 256 + threadIdx.x;
    if (i < N) {
        float d = deg[i];
        deg[i] = (d > 0.0f) ? rsqrtf(d) : 0.0f;
    }
}

__global__ __launch_bounds__(256) void edge_norm(const int* __restrict__ src,
                                                 const int* __restrict__ dst,
                                                 const float* __restrict__ dinv,
                                                 float* __restrict__ nrm, int E) {
    int e = blockIdx.x * 256 + threadIdx.x;
    if (e < E) nrm[e] = dinv[src[e]] * dinv[dst[e]];
}

// one wave (32 lanes) per row of 128 features; shuffle-reduce the row sum
__global__ __launch_bounds__(256) void row_normalize(const float* __restrict__ x,
                                                     float* __restrict__ h, int N) {
    int wave = (blockIdx.x * 256 + threadIdx.x) >> 5;
    int lane = threadIdx.x & 31;
    if (wave >= N) return;
    const float4 v = *(const float4*)(x + (size_t)wave * NFEAT_IN + lane * 4);
    float s = v.x + v.y + v.z + v.w;
    #pragma unroll
    for (int off = 16; off > 0; off >>= 1) s += __shfl_xor(s, off, 32);
    float inv = 1.0f / fmaxf(s, 1e-8f);
    float4 o;
    o.x = v.x * inv; o.y = v.y * inv; o.z = v.z * inv; o.w = v.w * inv;
    *(float4*)(h + (size_t)wave * NFEAT_IN + lane * 4) = o;
}

// ----------------------------- sparse propagation --------------------------
// out[dst] += norm[e] * in[src]; D/4 threads per edge, float4 gather,
// per-float global atomic add (L2 does the RMW; whole graph is L2-resident).
template <int D>
__global__ __launch_bounds__(256) void propagate(const float* __restrict__ Hin,
                                                 float* __restrict__ Hout,
                                                 const int* __restrict__ src,
                                                 const int* __restrict__ dst,
                                                 const float* __restrict__ nrm, int E) {
    const int CPE = D / 4;
    long long t = (long long)blockIdx.x * 256 + threadIdx.x;
    long long total = (long long)E * CPE;
    if (t >= total) return;
    int e = (int)(t / CPE);
    int c = (int)(t % CPE);
    float w = nrm[e];
    int s = src[e];
    int d = dst[e];
    const float4 v = *(const float4*)(Hin + (size_t)s * D + c * 4);
    float* o = Hout + (size_t)d * D + c * 4;
    atomicAdd(o + 0, w * v.x);
    atomicAdd(o + 1, w * v.y);
    atomicAdd(o + 2, w * v.z);
    atomicAdd(o + 3, w * v.w);
}

// ----------------------------- WMMA dense combine ---------------------------
// Y[tile] (+)= H_tile @ W_k using V_WMMA_F32_16X16X4_F32.
// 8 waves per block, one 16-row tile per wave. W_k staged in LDS.
// Fragment loads are software-pipelined one K-step ahead of the WMMA chain.
// pass 0: init accumulator from bias; last layer-1 pass applies ReLU.
template <int D_IN, int D_OUT>
__global__ __launch_bounds__(256) void tag_gemm(const float* __restrict__ H,
                                                const float* __restrict__ Wk,
                                                const float* __restrict__ bias,
                                                float* __restrict__ Y,
                                                int N, int init_bias, int do_relu) {
    __shared__ float Wlds[D_IN * D_OUT];
    for (int i = threadIdx.x; i < D_IN * D_OUT; i += 256) Wlds[i] = Wk[i];
    __syncthreads();

    const int lane = threadIdx.x & 31;
    const int wave = threadIdx.x >> 5;
    const int hf   = lane >> 4;    // lane half: selects K pair / M+8
    const int m16  = lane & 15;
    const int tile = blockIdx.x * 128 + wave * 16;

    int row = tile + m16;
    if (row >= N) row = N - 1;                    // clamp: keep EXEC all-ones
    const float* hrow = H + (size_t)row * D_IN;

    // small-K case: keep all A fragments resident across output ntiles
    v2f afr[D_IN <= 32 ? D_IN / 4 : 1];
    if constexpr (D_IN <= 32) {
        #pragma unroll
        for (int kk = 0; kk < D_IN; kk += 4)
            afr[kk / 4] = *(const v2f*)(hrow + kk + 2 * hf);
    }

    #pragma unroll
    for (int nt = 0; nt < D_OUT / 16; ++nt) {
        const int oc = nt * 16 + m16;
        const float* wcol = &Wlds[nt * 16 + m16];   // B column base in LDS

        v8f acc;
        if (init_bias) {
            float bv = bias[oc];
            #pragma unroll
            for (int r = 0; r < 8; ++r) acc[r] = bv;
        } else {
            #pragma unroll
            for (int r = 0; r < 8; ++r) {
                int rr = tile + r + 8 * hf;
                if (rr >= N) rr = N - 1;
                acc[r] = Y[(size_t)rr * D_OUT + oc];
            }
        }

        // prologue: fragments for kk = 0
        v2f a, b;
        if constexpr (D_IN <= 32) a = afr[0];
        else                      a = *(const v2f*)(hrow + 2 * hf);
        b.x = wcol[(2 * hf) * D_OUT];
        b.y = wcol[(2 * hf + 1) * D_OUT];

        #pragma unroll
        for (int kk = 0; kk < D_IN; kk += 4) {
            // prefetch next K-step before issuing the current WMMA
            v2f a_n = a, b_n = b;
            if (kk + 4 < D_IN) {
                if constexpr (D_IN <= 32) a_n = afr[kk / 4 + 1];
                else                      a_n = *(const v2f*)(hrow + kk + 4 + 2 * hf);
                b_n.x = wcol[(kk + 4 + 2 * hf) * D_OUT];
                b_n.y = wcol[(kk + 5 + 2 * hf) * D_OUT];
            }
            acc = __builtin_amdgcn_wmma_f32_16x16x4_f32(
                /*neg_a=*/false, a, /*neg_b=*/false, b,
                /*c_mod=*/(short)0, acc, /*reuse_a=*/false, /*reuse_b=*/false);
            a = a_n; b = b_n;
        }

        #pragma unroll
        for (int r = 0; r < 8; ++r) {
            int rr = tile + r + 8 * hf;
            if (rr < N) {
                float v = acc[r];
                if (do_relu) v = fmaxf(v, 0.0f);
                Y[(size_t)rr * D_OUT + oc] = v;
            }
        }
    }
}

// ----------------------------- launch sequence ------------------------------

extern "C" void kernel_launch(void* const* d_in, const int* in_sizes, int n_in,
                              void* d_out, int out_size, void* d_ws, size_t ws_size,
                              hipStream_t stream) {
    const float* x  = (const float*)d_in[0];
    const int*   ei = (const int*)d_in[1];
    const float* W1 = (const float*)d_in[2];
    const float* b1 = (const float*)d_in[3];
    const float* W2 = (const float*)d_in[4];
    const float* b2 = (const float*)d_in[5];
    float* out = (float*)d_out;

    const int N = in_sizes[0] / NFEAT_IN;   // 50000
    const int E = in_sizes[1] / 2;          // 800000
    const int* srcI = ei;                   // edge_index[0]
    const int* dstI = ei + E;               // edge_index[1]

    // workspace carve-up (floats), 64-elem aligned blocks; total ~64 MB
    float* ws   = (float*)d_ws;
    float* deg  = ws;
    float* nrm  = deg + (((size_t)N + 63) & ~(size_t)63);
    float* hopA = nrm + (((size_t)E + 63) & ~(size_t)63);
    float* hopB = hopA + (size_t)N * NFEAT_IN;
    float* h1   = hopB + (size_t)N * NFEAT_IN;   // layer-1 output [N,16]
    float* h2A  = h1   + (size_t)N * 16;
    float* h2B  = h2A  + (size_t)N * 16;

    const int T = 256;
    const int gN  = (N + T - 1) / T;
    const int gE  = (E + T - 1) / T;
    const int gRN = (N + 7) / 8;                 // row_normalize: 8 waves/block
    const int gP128 = (int)(((long long)E * 32 + T - 1) / T);
    const int gP16  = (int)(((long long)E * 4  + T - 1) / T);
    const int gG    = (N + 127) / 128;           // gemm: 128 rows/block
    const long long nf128 = (long long)N * NFEAT_IN;
    const long long nf16  = (long long)N * 16;

    // --- gcn_norm ---
    zero_f32<<<gN, T, 0, stream>>>(deg, N);
    deg_accum<<<gE, T, 0, stream>>>(dstI, deg, E);
    deg_inv_sqrt<<<gN, T, 0, stream>>>(deg, N);
    edge_norm<<<gE, T, 0, stream>>>(srcI, dstI, deg, nrm, E);

    // --- h = row_normalize(x) ---
    row_normalize<<<gRN, T, 0, stream>>>(x, hopA, N);

    // --- layer 1: accumulate hop-k @ W1[k] into h1, ReLU on last pass ---
    tag_gemm<128, 16><<<gG, T, 0, stream>>>(hopA, W1 + 0 * 128 * 16, b1, h1, N, 1, 0);

    zero_f32<<<(int)((nf128 + T - 1) / T), T, 0, stream>>>(hopB, nf128);
    propagate<128><<<gP128, T, 0, stream>>>(hopA, hopB, srcI, dstI, nrm, E);
    tag_gemm<128, 16><<<gG, T, 0, stream>>>(hopB, W1 + 1 * 128 * 16, b1, h1, N, 0, 0);

    zero_f32<<<(int)((nf128 + T - 1) / T), T, 0, stream>>>(hopA, nf128);
    propagate<128><<<gP128, T, 0, stream>>>(hopB, hopA, srcI, dstI, nrm, E);
    tag_gemm<128, 16><<<gG, T, 0, stream>>>(hopA, W1 + 2 * 128 * 16, b1, h1, N, 0, 0);

    zero_f32<<<(int)((nf128 + T - 1) / T), T, 0, stream>>>(hopB, nf128);
    propagate<128><<<gP128, T, 0, stream>>>(hopA, hopB, srcI, dstI, nrm, E);
    tag_gemm<128, 16><<<gG, T, 0, stream>>>(hopB, W1 + 3 * 128 * 16, b1, h1, N, 0, 1);

    // --- layer 2: accumulate hop-k @ W2[k] into d_out ---
    tag_gemm<16, 64><<<gG, T, 0, stream>>>(h1, W2 + 0 * 16 * 64, b2, out, N, 1, 0);

    zero_f32<<<(int)((nf16 + T - 1) / T), T, 0, stream>>>(h2B, nf16);
    propagate<16><<<gP16, T, 0, stream>>>(h1, h2B, srcI, dstI, nrm, E);
    tag_gemm<16, 64><<<gG, T, 0, stream>>>(h2B, W2 + 1 * 16 * 64, b2, out, N, 0, 0);

    zero_f32<<<(int)((nf16 + T - 1) / T), T, 0, stream>>>(h2A, nf16);
    propagate<16><<<gP16, T, 0, stream>>>(h2B, h2A, srcI, dstI, nrm, E);
    tag_gemm<16, 64><<<gG, T, 0, stream>>>(h2A, W2 + 2 * 16 * 64, b2, out, N, 0, 0);

    zero_f32<<<(int)((nf16 + T - 1) / T), T, 0, stream>>>(h2B, nf16);
    propagate<16><<<gP16, T, 0, stream>>>(h2A, h2B, srcI, dstI, nrm, E);
    tag_gemm<16, 64><<<gG, T, 0, stream>>>(h2B, W2 + 3 * 16 * 64, b2, out, N, 0, 0);
}